// AdaptedCrossEntropySurvivalLoss_76785425318337
// MI455X (gfx1250) — compile-verified
//
#include <hip/hip_runtime.h>

typedef __attribute__((ext_vector_type(2))) float v2f;
typedef __attribute__((ext_vector_type(8))) float v8f;

#define WPB   8     // waves per 256-thread block (wave32)
#define TBINS 64
#define LN2F  0.69314718055994530942f

// One wave processes a tile of 16 rows.
// lane = 16*half + mrow : lane handles cols [32*half, 32*half+32) of row (tilebase + mrow).
// Per-lane partials are log2-sums; a single V_WMMA_F32_16X16X4_F32 against an
// all-ones B matrix reduces the 16 rows at once.
__global__ __launch_bounds__(256) void survival_loss_main(
    const float* __restrict__ preds,
    const long long* __restrict__ targets,
    float* __restrict__ partials,
    int n)
{
  const int lane = threadIdx.x & 31;
  const int wave = threadIdx.x >> 5;
  const int mrow = lane & 15;    // row within tile
  const int half = lane >> 4;    // 0: cols 0-31, 1: cols 32-63
  const int tiles = (n + 15) >> 4;

  float wave_acc = 0.0f;

  for (int tile = blockIdx.x * WPB + wave; tile < tiles; tile += gridDim.x * WPB) {
    const int r = (tile << 4) + mrow;

    // Lanes 0-15 fetch the 16B target pair once per row; broadcast to upper half.
    int packed = 0;
    if (lane < 16 && r < n) {
      int t = (int)targets[2 * (long long)r];
      t = t < 0 ? 0 : (t > TBINS - 1 ? TBINS - 1 : t);
      const int ev = targets[2 * (long long)r + 1] != 0 ? 1 : 0;
      packed = t | (ev << 8);
    }
    packed = __shfl(packed, mrow, 32);
    const int  t  = packed & 0xFF;
    const bool ev = (packed >> 8) != 0;

    float partial = 0.0f;   // sum of log2(x) for this half-row
    const int jbase = half << 5;

    if (r < n && jbase <= t) {   // skip the second 128B cacheline when t < 32
      const float4* rowp = (const float4*)(preds + (size_t)r * TBINS + jbase);
      float4 p[8];
      #pragma unroll
      for (int m = 0; m < 8; ++m) p[m] = rowp[m];   // 8x global_load_b128, one wait

      #pragma unroll
      for (int m = 0; m < 8; ++m) {
        float pv[4] = {p[m].x, p[m].y, p[m].z, p[m].w};
        #pragma unroll
        for (int c = 0; c < 4; ++c) {
          const int j = jbase + (m << 2) + c;
          float h  = fminf(fmaxf(pv[c], 1e-9f), 1.0f - 1e-9f);
          float om = 1.0f - h;
          // x: j<t -> 1-h ; j==t -> (ev ? h : 1-h) ; j>t -> 1 (log2=0)
          float x = (j < t) ? om : 1.0f;
          x = (j == t) ? (ev ? h : om) : x;
          partial += __builtin_amdgcn_logf(x);   // v_log_f32 (log2), 1 instr
        }
      }
    }

    // A[M][0] = half0 partial of row M, A[M][2] = half1 partial, A[M][1]=A[M][3]=0.
    // B = all-ones 4x16  =>  D[M][n] = row-M total in every column n.
    v2f A; A.x = partial; A.y = 0.0f;
    v2f B; B.x = 1.0f;    B.y = 1.0f;
    v8f C = {};
    C = __builtin_amdgcn_wmma_f32_16x16x4_f32(
        /*neg_a=*/false, A, /*neg_b=*/false, B,
        /*c_mod=*/(short)0, C, /*reuse_a=*/false, /*reuse_b=*/false);

    float s8 = C[0] + C[1] + C[2] + C[3] + C[4] + C[5] + C[6] + C[7];
    // lane 0 column: rows 0-7 total; lane 16 column: rows 8-15 total
    wave_acc += __shfl(s8, 0, 32) + __shfl(s8, 16, 32);
  }

  __shared__ float wacc[WPB];
  if (lane == 0) wacc[wave] = wave_acc;
  __syncthreads();
  if (threadIdx.x == 0) {
    float s = 0.0f;
    #pragma unroll
    for (int i = 0; i < WPB; ++i) s += wacc[i];
    partials[blockIdx.x] = s;   // deterministic ordered block partial (log2 units)
  }
}

__global__ __launch_bounds__(256) void survival_loss_reduce(
    const float* __restrict__ partials, int nparts,
    float* __restrict__ out, float inv_n)
{
  __shared__ float buf[256];
  float s = 0.0f;
  for (int i = threadIdx.x; i < nparts; i += 256) s += partials[i];
  buf[threadIdx.x] = s;
  __syncthreads();
  for (int off = 128; off > 0; off >>= 1) {
    if ((int)threadIdx.x < off) buf[threadIdx.x] += buf[threadIdx.x + off];
    __syncthreads();
  }
  // loss = -ln2 * sum(log2 terms) / N
  if (threadIdx.x == 0) out[0] = -buf[0] * inv_n * LN2F;
}

extern "C" void kernel_launch(void* const* d_in, const int* in_sizes, int n_in,
                              void* d_out, int out_size, void* d_ws, size_t ws_size,
                              hipStream_t stream) {
  const float*     preds   = (const float*)d_in[0];
  const long long* targets = (const long long*)d_in[1];   // int64 (N,2)
  float* out      = (float*)d_out;
  float* partials = (float*)d_ws;

  const int n = in_sizes[0] / TBINS;   // N rows

  int nblocks = 2048;
  size_t maxp = ws_size / sizeof(float);
  if ((size_t)nblocks > maxp) nblocks = (int)maxp;
  if (nblocks < 1) nblocks = 1;

  survival_loss_main<<<nblocks, 256, 0, stream>>>(preds, targets, partials, n);
  survival_loss_reduce<<<1, 256, 0, stream>>>(partials, nblocks, out, 1.0f / (float)n);
}